// AttentionIntegrator_9723805958439
// MI455X (gfx1250) — compile-verified
//
#include <hip/hip_runtime.h>

typedef float v2f __attribute__((ext_vector_type(2)));
typedef float v8f __attribute__((ext_vector_type(8)));
typedef int   v4i __attribute__((ext_vector_type(4)));

static constexpr int Mtot = 32768;   // B*V rows
static constexpr int Kdim = 1024;    // D
static constexpr int Ndim = 1024;    // H
static constexpr int NB   = 8192;    // samples
static constexpr int NC   = 10;      // classes

#define AS_GLOBAL __attribute__((address_space(1)))
#define AS_LDS    __attribute__((address_space(3)))

// 16-byte global -> LDS async copy (ASYNCcnt-tracked, no VGPR staging).
__device__ __forceinline__ void async_copy_b128(const float* g, float* l) {
#if __has_builtin(__builtin_amdgcn_global_load_async_to_lds_b128)
  float* gm = const_cast<float*>(g);
  AS_GLOBAL v4i* gp = (AS_GLOBAL v4i*)gm;
  AS_LDS    v4i* lp = (AS_LDS v4i*)l;
  __builtin_amdgcn_global_load_async_to_lds_b128(gp, lp, 0, 0);
#else
  const unsigned lds_addr = (unsigned)(__UINTPTR_TYPE__)(AS_LDS float*)l;
  asm volatile("global_load_async_to_lds_b128 %0, %1, off"
               :
               : "v"(lds_addr), "v"(g)
               : "memory");
#endif
}

template <int N>
__device__ __forceinline__ void wait_asynccnt() {
#if __has_builtin(__builtin_amdgcn_s_wait_asynccnt)
  __builtin_amdgcn_s_wait_asynccnt(N);
#else
  asm volatile("s_wait_asynccnt %0" ::"n"(N) : "memory");
#endif
}

// ---------------------------------------------------------------------------
// Tiled fp32 WMMA GEMM: C[m] = relu?(A @ B[m] + bias[m])   for m in [0,NMAT)
// Block tile 64(M) x 32(N), K-step 32; 256 threads = 8 waves, each wave owns
// a 16x16 tile per output matrix (V_WMMA_F32_16X16X4_F32, fp32-exact).
// Double-buffered LDS filled by async global->LDS copies so tile i+1 streams
// in while tile i is consumed by the matrix pipe.
// ---------------------------------------------------------------------------
template <int NMAT, bool RELU>
__global__ __launch_bounds__(256) void gemm_wmma_f32(
    const float* __restrict__ A,
    const float* __restrict__ B0, const float* __restrict__ B1,
    const float* __restrict__ B2,
    const float* __restrict__ bias0, const float* __restrict__ bias1,
    const float* __restrict__ bias2,
    float* __restrict__ C0, float* __restrict__ C1, float* __restrict__ C2) {
  // 36-float row stride: keeps every 16B async store aligned (144B rows) and
  // is bank-conflict-free for row-strided A reads and column reads of B.
  __shared__ float As[2][64][36];
  __shared__ float Bs[2][NMAT][32][36];

  constexpr int TILES = Kdim / 32;       // 32 K-tiles
  constexpr int PT    = 2 + NMAT;        // async copies per thread per tile

  const int t     = threadIdx.x;
  const int mBase = blockIdx.x * 64;
  const int nBase = blockIdx.y * 32;
  const int lane  = t & 31;
  const int wave  = t >> 5;
  const int wm    = wave & 3;            // wave's M-tile (0..3)
  const int wn    = wave >> 2;           // wave's N-tile (0..1)
  const int lr    = lane & 15;
  const int half  = lane >> 4;           // 0: K=0,1 / rows 0-7 ; 1: K=2,3 / rows 8-15

  const float* Bp[3]    = {B0, B1, B2};
  const float* biasp[3] = {bias0, bias1, bias2};
  float*       Cp[3]    = {C0, C1, C2};

  // Per-thread fixed copy slots.
  const int arow0 = t >> 2;              // A slot: rows 0..63
  const int ac40  = (t & 3) << 3;        //   two b128 each: cols 0,8,16,24 (+4)
  const int brow  = t >> 3;              // B slot: rows 0..31
  const int bc4   = (t & 7) << 2;

  auto issue_tile = [&](int kt, int buf) {
    const int k0 = kt * 32;
    // A tile 64x32 : 512 b128 slots -> 2 per thread
    async_copy_b128(A + (size_t)(mBase + arow0) * Kdim + k0 + ac40,
                    &As[buf][arow0][ac40]);
    async_copy_b128(A + (size_t)(mBase + arow0) * Kdim + k0 + ac40 + 4,
                    &As[buf][arow0][ac40 + 4]);
    // B tiles 32x32 each : 256 b128 slots -> 1 per thread per matrix
#pragma unroll
    for (int m = 0; m < NMAT; ++m)
      async_copy_b128(Bp[m] + (size_t)(k0 + brow) * Ndim + nBase + bc4,
                      &Bs[buf][m][brow][bc4]);
  };

  v8f acc[NMAT];
#pragma unroll
  for (int m = 0; m < NMAT; ++m)
    acc[m] = (v8f){0.f, 0.f, 0.f, 0.f, 0.f, 0.f, 0.f, 0.f};

  issue_tile(0, 0);

  for (int i = 0; i < TILES; ++i) {
    const int buf = i & 1;
    if (i + 1 < TILES) {
      issue_tile(i + 1, buf ^ 1);        // prefetch next tile into other buffer
      wait_asynccnt<PT>();               // tile i landed; tile i+1 in flight
    } else {
      wait_asynccnt<0>();
    }
    __syncthreads();

#pragma unroll
    for (int kk = 0; kk < 32; kk += 4) {
      // A frag 16x4: lanes 0-15 carry K=kk,kk+1; lanes 16-31 carry K=kk+2,kk+3
      v2f a;
      a.x = As[buf][wm * 16 + lr][kk + half * 2 + 0];
      a.y = As[buf][wm * 16 + lr][kk + half * 2 + 1];
#pragma unroll
      for (int m = 0; m < NMAT; ++m) {
        v2f b;
        b.x = Bs[buf][m][kk + half * 2 + 0][wn * 16 + lr];
        b.y = Bs[buf][m][kk + half * 2 + 1][wn * 16 + lr];
        acc[m] = __builtin_amdgcn_wmma_f32_16x16x4_f32(
            false, a, false, b, (short)0, acc[m], false, false);
      }
    }
    __syncthreads();  // all waves done reading buf before it is refilled
  }

  // Epilogue: C layout -> VGPR r holds row half*8+r, lane%16 is the column.
  const int row0 = mBase + wm * 16 + half * 8;
  const int col  = nBase + wn * 16 + lr;
#pragma unroll
  for (int m = 0; m < NMAT; ++m) {
    const float bv = biasp[m][col];
#pragma unroll
    for (int r = 0; r < 8; ++r) {
      float o = acc[m][r] + bv;
      if (RELU) o = fmaxf(o, 0.f);
      Cp[m][(size_t)(row0 + r) * Ndim + col] = o;
    }
  }
}

// ---------------------------------------------------------------------------
// Per-sample attention over V=4 + residual + LayerNorm over (V*H)=4096 elems.
// One 256-thread block per sample; each thread owns 4 h-positions per view.
// xn may alias q (reads of q complete, barrier, then writes).
// ---------------------------------------------------------------------------
__global__ __launch_bounds__(256) void attn_ln_kernel(
    const float* q, const float* k, const float* v,
    const float* __restrict__ xt, float* xn) {
  const int b    = blockIdx.x;
  const int t    = threadIdx.x;
  const int lane = t & 31;
  const int wave = t >> 5;
  const size_t base = (size_t)b * 4 * 1024;

  float qv[4][4], kv[4][4], vv[4][4], xv[4][4];
#pragma unroll
  for (int vi = 0; vi < 4; ++vi)
#pragma unroll
    for (int j = 0; j < 4; ++j) {
      const size_t idx = base + (size_t)vi * 1024 + (j * 256 + t);
      qv[vi][j] = q[idx];
      kv[vi][j] = k[idx];
      vv[vi][j] = v[idx];
      xv[vi][j] = xt[idx];
    }

  // 16 partial score dot-products
  float s[16];
#pragma unroll
  for (int i = 0; i < 16; ++i) s[i] = 0.f;
#pragma unroll
  for (int vi = 0; vi < 4; ++vi)
#pragma unroll
    for (int wi = 0; wi < 4; ++wi)
#pragma unroll
      for (int j = 0; j < 4; ++j) s[vi * 4 + wi] += qv[vi][j] * kv[wi][j];

  // wave reduction, then cross-wave via LDS
#pragma unroll
  for (int off = 16; off >= 1; off >>= 1)
#pragma unroll
    for (int i = 0; i < 16; ++i) s[i] += __shfl_xor(s[i], off, 32);

  __shared__ float red[8 * 16];
  __shared__ float sc[16];
  __shared__ float red2[16];
  __shared__ float stats[2];
  if (lane == 0)
#pragma unroll
    for (int i = 0; i < 16; ++i) red[wave * 16 + i] = s[i];
  __syncthreads();
  if (t < 16) {
    float tot = 0.f;
#pragma unroll
    for (int w = 0; w < 8; ++w) tot += red[w * 16 + t];
    sc[t] = tot * 0.03125f;  // 1/sqrt(H), H=1024
  }
  __syncthreads();

  // softmax rows (each thread redundantly; 4x4 is trivial)
  float attn[4][4];
#pragma unroll
  for (int vi = 0; vi < 4; ++vi) {
    float mx = sc[vi * 4 + 0];
#pragma unroll
    for (int w = 1; w < 4; ++w) mx = fmaxf(mx, sc[vi * 4 + w]);
    float sum = 0.f;
#pragma unroll
    for (int w = 0; w < 4; ++w) {
      attn[vi][w] = __expf(sc[vi * 4 + w] - mx);
      sum += attn[vi][w];
    }
    const float inv = 1.f / sum;
#pragma unroll
    for (int w = 0; w < 4; ++w) attn[vi][w] *= inv;
  }

  // x = attn @ v + xt ; accumulate LN partials
  float x[4][4];
  float sum = 0.f, sq = 0.f;
#pragma unroll
  for (int vi = 0; vi < 4; ++vi)
#pragma unroll
    for (int j = 0; j < 4; ++j) {
      float a = xv[vi][j];
#pragma unroll
      for (int w = 0; w < 4; ++w) a += attn[vi][w] * vv[w][j];
      x[vi][j] = a;
      sum += a;
      sq += a * a;
    }
#pragma unroll
  for (int off = 16; off >= 1; off >>= 1) {
    sum += __shfl_xor(sum, off, 32);
    sq += __shfl_xor(sq, off, 32);
  }
  if (lane == 0) {
    red2[wave] = sum;
    red2[8 + wave] = sq;
  }
  __syncthreads();
  if (t == 0) {
    float S = 0.f, Q = 0.f;
#pragma unroll
    for (int w = 0; w < 8; ++w) {
      S += red2[w];
      Q += red2[8 + w];
    }
    const float mean = S * (1.f / 4096.f);
    const float var  = Q * (1.f / 4096.f) - mean * mean;
    stats[0] = mean;
    stats[1] = rsqrtf(var + 1e-5f);
  }
  __syncthreads();
  const float mean = stats[0], rstd = stats[1];
#pragma unroll
  for (int vi = 0; vi < 4; ++vi)
#pragma unroll
    for (int j = 0; j < 4; ++j) {
      const size_t idx = base + (size_t)vi * 1024 + (j * 256 + t);
      xn[idx] = (x[vi][j] - mean) * rstd;
    }
}

// ---------------------------------------------------------------------------
// Classifier: out(8192x10) = h(8192x4096) @ Wf(4096x10) + bf
// ---------------------------------------------------------------------------
__global__ __launch_bounds__(256) void head_kernel(
    const float* __restrict__ h, const float* __restrict__ Wf,
    const float* __restrict__ bf, float* __restrict__ out) {
  const int b    = blockIdx.x;
  const int t    = threadIdx.x;
  const int lane = t & 31;
  const int wave = t >> 5;
  const float* xr = h + (size_t)b * 4096;

  float acc[NC];
#pragma unroll
  for (int c = 0; c < NC; ++c) acc[c] = 0.f;
  for (int j = t; j < 4096; j += 256) {
    const float xv = xr[j];
    const float* wr = Wf + (size_t)j * NC;
#pragma unroll
    for (int c = 0; c < NC; ++c) acc[c] += xv * wr[c];
  }
#pragma unroll
  for (int off = 16; off >= 1; off >>= 1)
#pragma unroll
    for (int c = 0; c < NC; ++c) acc[c] += __shfl_xor(acc[c], off, 32);

  __shared__ float red[8][NC];
  if (lane == 0)
#pragma unroll
    for (int c = 0; c < NC; ++c) red[wave][c] = acc[c];
  __syncthreads();
  if (t < NC) {
    float tot = bf[t];
#pragma unroll
    for (int w = 0; w < 8; ++w) tot += red[w][t];
    out[(size_t)b * NC + t] = tot;
  }
}

// ---------------------------------------------------------------------------
extern "C" void kernel_launch(void* const* d_in, const int* in_sizes, int n_in,
                              void* d_out, int out_size, void* d_ws,
                              size_t ws_size, hipStream_t stream) {
  const float* xt = (const float*)d_in[0];
  const float* Wq = (const float*)d_in[1];
  const float* bq = (const float*)d_in[2];
  const float* Wk = (const float*)d_in[3];
  const float* bk = (const float*)d_in[4];
  const float* Wv = (const float*)d_in[5];
  const float* bv = (const float*)d_in[6];
  const float* W1 = (const float*)d_in[7];
  const float* b1 = (const float*)d_in[8];
  const float* Wf = (const float*)d_in[9];
  const float* bf = (const float*)d_in[10];

  const size_t mat = (size_t)Mtot * Ndim;  // 32768*1024 floats = 128 MiB
  float* q  = (float*)d_ws;                // region 0
  float* k  = q + mat;                     // region 1
  float* v  = k + mat;                     // region 2
  float* xn = q;                           // reuse region 0 (q dead after K2)
  float* hh = k;                           // reuse region 1 (k dead after K2)

  dim3 ggrid(Mtot / 64, Ndim / 32);        // 512 x 32 blocks

  // K1: fused QKV projection (3 GEMMs sharing A-tile async loads)
  gemm_wmma_f32<3, false><<<ggrid, 256, 0, stream>>>(
      xt, Wq, Wk, Wv, bq, bk, bv, q, k, v);

  // K2: attention over V=4 + residual + LayerNorm
  attn_ln_kernel<<<NB, 256, 0, stream>>>(q, k, v, xt, xn);

  // K3: FFN GEMM + bias + ReLU
  gemm_wmma_f32<1, true><<<ggrid, 256, 0, stream>>>(
      xn, W1, W1, W1, b1, b1, b1, hh, hh, hh);

  // K4: classifier head
  head_kernel<<<NB, 256, 0, stream>>>(hh, Wf, bf, (float*)d_out);
}